// Game_30296699306253
// MI455X (gfx1250) — compile-verified
//
#include <hip/hip_runtime.h>
#include <stdint.h>

#define A_NUM    512
#define T_END    8
#define FC1      128
#define NSTEP    511         // A_NUM - 1 trade steps per round
#define NTHREADS 512         // 16 wave32s, one WGP

typedef __attribute__((ext_vector_type(2))) float v2f;
typedef __attribute__((ext_vector_type(8))) float v8f;

// ---------------- JAX threefry2x32 (exact) ----------------
__device__ __forceinline__ uint32_t rotl32(uint32_t v, int r) {
    return (v << r) | (v >> (32 - r));
}

__device__ __forceinline__ void threefry2x32(uint32_t k0, uint32_t k1,
                                             uint32_t x0, uint32_t x1,
                                             uint32_t& o0, uint32_t& o1) {
    const uint32_t ks0 = k0, ks1 = k1, ks2 = k0 ^ k1 ^ 0x1BD11BDAu;
    x0 += ks0; x1 += ks1;
    const int rA[4] = {13, 15, 26, 6};
    const int rB[4] = {17, 29, 16, 24};
#define TF_R4(R)                                                   \
    { _Pragma("unroll") for (int j = 0; j < 4; ++j) {              \
          x0 += x1; x1 = rotl32(x1, (R)[j]); x1 ^= x0; } }
    TF_R4(rA); x0 += ks1; x1 += ks2 + 1u;
    TF_R4(rB); x0 += ks2; x1 += ks0 + 2u;
    TF_R4(rA); x0 += ks0; x1 += ks1 + 3u;
    TF_R4(rB); x0 += ks1; x1 += ks2 + 4u;
    TF_R4(rA); x0 += ks2; x1 += ks0 + 5u;
#undef TF_R4
    o0 = x0; o1 = x1;
}

// rsel[i] for round t: r = min(int(u_i * (512-i)), 511-i), u_i from
// uniform(split(fold_in(key(42), t), 511)[i]).  Each i is independent:
// keys[i] = (concat[2i], concat[2i+1]) with concat = [out0 ; out1] of
// threefry(fk, j, 511+j), j = 0..510.
__device__ __forceinline__ void compute_rsel(int t, int* dst, int j0, int stride) {
    uint32_t fk0, fk1;
    threefry2x32(0u, 42u, 0u, (uint32_t)t, fk0, fk1);   // fold_in(key(42), t)
    for (int i = j0; i < NSTEP; i += stride) {
        uint32_t c0, c1, o0, o1, d0, d1;
        if (2 * i + 1 < NSTEP) {                        // i <= 254: both in out0
            threefry2x32(fk0, fk1, (uint32_t)(2 * i),     (uint32_t)(NSTEP + 2 * i),     o0, o1); c0 = o0;
            threefry2x32(fk0, fk1, (uint32_t)(2 * i + 1), (uint32_t)(NSTEP + 2 * i + 1), d0, d1); c1 = d0;
        } else if (2 * i == NSTEP - 1) {                // i == 255: straddle
            threefry2x32(fk0, fk1, (uint32_t)(NSTEP - 1), (uint32_t)(2 * NSTEP - 1), o0, o1); c0 = o0;
            threefry2x32(fk0, fk1, 0u,                    (uint32_t)NSTEP,           d0, d1); c1 = d1;
        } else {                                        // i >= 256: both in out1
            threefry2x32(fk0, fk1, (uint32_t)(2 * i - NSTEP),     (uint32_t)(2 * i),     o0, o1); c0 = o1;
            threefry2x32(fk0, fk1, (uint32_t)(2 * i + 1 - NSTEP), (uint32_t)(2 * i + 1), d0, d1); c1 = d1;
        }
        uint32_t u0, u1;
        threefry2x32(c0, c1, 0u, 0u, u0, u1);           // uniform(keys[i])
        float u = __uint_as_float((u0 >> 9) | 0x3f800000u) - 1.0f;
        int na = A_NUM - i;
        dst[i] = min((int)(u * (float)na), na - 1);
    }
}

// ---------------- one-wave price of one agent (fallback path) ----------------
__device__ __forceinline__ float wave_price(int a, float ret, int lane,
                                            const float* __restrict__ W1,
                                            const float* __restrict__ b1,
                                            const float* __restrict__ W2,
                                            const float* __restrict__ b2) {
    const float* w1p = W1 + a * FC1;
    const float* b1p = b1 + a * FC1;
    const float* w2p = W2 + a * FC1;
    float acc = 0.0f;
#pragma unroll
    for (int h = lane; h < FC1; h += 32) {
        float hv = fmaxf(fmaf(ret, w1p[h], b1p[h]), 0.0f);
        acc = fmaf(hv, w2p[h], acc);
    }
#pragma unroll
    for (int off = 16; off > 0; off >>= 1)
        acc += __shfl_xor(acc, off, 32);
    return acc + b2[a];
}

__global__ __launch_bounds__(NTHREADS)
void market_sim_kernel(const float* __restrict__ x,
                       const float* __restrict__ W1,
                       const float* __restrict__ b1,
                       const float* __restrict__ W2,
                       const float* __restrict__ b2,
                       float* __restrict__ out) {
    __shared__ float    book[2 * A_NUM];    // cash[512], stock[512]
    __shared__ float    mprice[A_NUM];      // price if active, +inf if not
    __shared__ uint32_t amask[16];          // active bitmask (wave 0 only)
    __shared__ int      rsel[2][NSTEP];     // double-buffered RNG ranks

    const int tid  = threadIdx.x;
    const int lane = tid & 31;
    const int wv   = tid >> 5;

    // book0 = x[16:]
    for (int i = tid; i < 2 * A_NUM; i += NTHREADS) book[i] = x[2 * T_END + i];
    // RNG table for round 0 (all threads)
    compute_rsel(0, rsel[0], tid, NTHREADS);
    __syncthreads();

    for (int t = 0; t < T_END; ++t) {
        const float dt = x[t];
        const float dT = x[T_END + t];

        // ---- full price recompute at round start (all 16 waves) ----
#if __has_builtin(__builtin_amdgcn_wmma_f32_16x16x4_f32)
        // 32 groups of 16 agents; D(16x16) = sum_k A(16x4)*B(4x16); diag = prices
        for (int g = wv; g < 32; g += 16) {
            const int m = lane & 15;
            const int a = g * 16 + m;
            const float ret = dt * book[a] + dT * book[A_NUM + a];
            const int klo = (lane >> 4) * 2;    // lanes 0-15: K{0,1}; 16-31: K{2,3}
            const float* w1p = W1 + a * FC1;
            const float* b1p = b1 + a * FC1;
            const float* w2p = W2 + a * FC1;    // B col N == lane&15 == same agent
            v8f c = {};
#pragma unroll 4
            for (int k = 0; k < 32; ++k) {
                const int h = 4 * k + klo;
                v2f av, bv;
                av.x = fmaxf(fmaf(ret, w1p[h],     b1p[h]),     0.0f);
                av.y = fmaxf(fmaf(ret, w1p[h + 1], b1p[h + 1]), 0.0f);
                bv.x = w2p[h];
                bv.y = w2p[h + 1];
                c = __builtin_amdgcn_wmma_f32_16x16x4_f32(
                        false, av, false, bv, (short)0, c, false, false);
            }
            // diagonal: m<8 at lane m (c[m]); m>=8 at lane m+16 (c[m-8])
            if (lane < 8) {
                int aa = g * 16 + lane;
                mprice[aa] = c[lane & 7] + b2[aa];
            } else if (lane >= 24) {
                int aa = g * 16 + (lane - 16);
                mprice[aa] = c[lane & 7] + b2[aa];
            }
        }
#else
        for (int a = wv * 32; a < wv * 32 + 32; ++a) {
            float ret = dt * book[a] + dT * book[A_NUM + a];
            float p = wave_price(a, ret, lane, W1, b1, W2, b2);
            if (lane == 0) mprice[a] = p;
        }
#endif
        __syncthreads();

        if (wv == 0) {
            // ================= wave 0: the 511 sequential steps, barrier-free ====
            const int* rs = rsel[t & 1];
            if (lane < 16) amask[lane] = 0xFFFFFFFFu;

            for (int s = 0; s < NSTEP; ++s) {
                // ---- bid = r-th active agent (lane 0), broadcast ----
                int bid = 0; float bid_p = 0.0f;
                if (lane == 0) {
                    int r = rs[s];
                    int w = 0; uint32_t mw = 0;
                    for (; w < 16; ++w) {
                        mw = amask[w];
                        int c = __popc(mw);
                        if (r < c) break;
                        r -= c;
                    }
                    for (int j = 0; j < r; ++j) mw &= (mw - 1u);
                    int bit = __ffs(mw) - 1;
                    bid = w * 32 + bit;
                    bid_p = mprice[bid];            // read before masking out
                    amask[w] &= ~(1u << bit);
                    mprice[bid] = __builtin_inff(); // bid leaves the market
                }
                bid   = __shfl(bid, 0, 32);
                bid_p = __shfl(bid_p, 0, 32);

                // ---- ask = first-index argmin over mprice (16 chunks/lane) ----
                float p = mprice[lane];
                int idx = lane;
#pragma unroll
                for (int c = 1; c < 16; ++c) {
                    float pc = mprice[lane + 32 * c];
                    if (pc < p) { p = pc; idx = lane + 32 * c; }
                }
#pragma unroll
                for (int off = 16; off > 0; off >>= 1) {
                    float po = __shfl_xor(p, off, 32);
                    int   io = __shfl_xor(idx, off, 32);
                    if (po < p || (po == p && io < idx)) { p = po; idx = io; }
                }
                const int   ask   = idx;    // uniform across lanes
                const float ask_p = p;

                // ---- prefetch ask's weights (hide L2 latency behind trade logic)
                const int h4 = 4 * lane;
                float4 w1v = *(const float4*)(W1 + ask * FC1 + h4);
                float4 b1v = *(const float4*)(b1 + ask * FC1 + h4);
                float4 w2v = *(const float4*)(W2 + ask * FC1 + h4);
                float  b2v = b2[ask];

                // ---- trade condition (uniform: all from broadcast values) ----
                const float cash      = book[bid];
                const float stocks    = book[A_NUM + ask];
                const float askCashO  = book[ask];
                const float bidStockO = book[A_NUM + bid];
                const bool cond = (bid_p >= ask_p) && (cash >= ask_p) && (stocks > 1.0f);

                if (cond) {
                    const float safe_p = (ask_p != 0.0f) ? ask_p : 1.0f;
                    const float q = fminf(floorf(cash / safe_p), stocks);
                    const float askCashN  = askCashO + ask_p;
                    const float askStockN = stocks + q;
                    if (lane == 0) {
                        book[bid]         = cash - ask_p;
                        book[ask]         = askCashN;
                        book[A_NUM + bid] = bidStockO + q;
                        book[A_NUM + ask] = askStockN;
                    }
                    // ---- refresh only the ask price (bid is inactive forever) ----
                    const float ret = dt * askCashN + dT * askStockN;
                    float acc;
                    acc = fmaxf(fmaf(ret, w1v.x, b1v.x), 0.0f) * w2v.x;
                    acc = fmaf(fmaxf(fmaf(ret, w1v.y, b1v.y), 0.0f), w2v.y, acc);
                    acc = fmaf(fmaxf(fmaf(ret, w1v.z, b1v.z), 0.0f), w2v.z, acc);
                    acc = fmaf(fmaxf(fmaf(ret, w1v.w, b1v.w), 0.0f), w2v.w, acc);
#pragma unroll
                    for (int off = 16; off > 0; off >>= 1)
                        acc += __shfl_xor(acc, off, 32);
                    if (lane == 0) mprice[ask] = acc + b2v;
                }
            }
        } else if (t + 1 < T_END) {
            // ===== waves 1..15: overlap next round's RNG table (independent) ====
            compute_rsel(t + 1, rsel[(t + 1) & 1], tid - 32, NTHREADS - 32);
        }
        __syncthreads();

        // ---- end of round: cash += returns ----
        {
            float cash = book[tid], stock = book[A_NUM + tid];
            book[tid] = cash + (dt * cash + dT * stock);
        }
        __syncthreads();
    }

    // ---- final payout ----
    const float r    = x[T_END - 1];
    const float d    = x[2 * T_END - 1];
    const float rate = d * (1.0f + r) / r;
    out[tid] = book[tid] + rate * book[A_NUM + tid];
}

extern "C" void kernel_launch(void* const* d_in, const int* in_sizes, int n_in,
                              void* d_out, int out_size, void* d_ws, size_t ws_size,
                              hipStream_t stream) {
    (void)in_sizes; (void)n_in; (void)d_ws; (void)ws_size; (void)out_size;
    const float* x  = (const float*)d_in[0];
    const float* W1 = (const float*)d_in[1];
    const float* b1 = (const float*)d_in[2];
    const float* W2 = (const float*)d_in[3];
    const float* b2 = (const float*)d_in[4];
    float* out = (float*)d_out;
    market_sim_kernel<<<1, NTHREADS, 0, stream>>>(x, W1, b1, W2, b2, out);
}